// So3kratesRepresentation_31559419691384
// MI455X (gfx1250) — compile-verified
//
#include <hip/hip_runtime.h>
#include <math.h>

typedef __attribute__((ext_vector_type(2))) float v2f;
typedef __attribute__((ext_vector_type(8))) float v8f;

#define R_MAX_F   6.0f
#define NUM_RADIAL 16
#define F_SCALAR  128
#define F_OUT     160
#define PI_F      3.14159265358979323846f
#define SQRT_2_OVER_RMAX 0.5773502691896258f   /* sqrt(2/6) */

#define S3    1.7320508075688772f
#define S15   3.872983346207417f
#define S10_4 0.7905694150420949f
#define S6_4  0.6123724356957945f
#define S15_2 1.9364916731037085f

// Coef[k][n]: sph[n] = sum_k mono[k] * Coef[k][n]. Monomial order k=0..19:
// 1, y, z, x, xy, yz, z2, xz, x2, y2, x2y, y3, xyz, yz2, z3, xz2, x2z, y2z, x3, xy2
__device__ __constant__ float SPH_COEF[20][16] = {
  /*k=0  1   */ {1.f,0,0,0, 0,0,-0.5f,0, 0,0,0,0, 0,0,0,0},
  /*k=1  y   */ {0,1.f,0,0, 0,0,0,0, 0,0,0,-S6_4, 0,0,0,0},
  /*k=2  z   */ {0,0,1.f,0, 0,0,0,0, 0,0,0,0, -1.5f,0,0,0},
  /*k=3  x   */ {0,0,0,1.f, 0,0,0,0, 0,0,0,0, 0,-S6_4,0,0},
  /*k=4  xy  */ {0,0,0,0, S3,0,0,0, 0,0,0,0, 0,0,0,0},
  /*k=5  yz  */ {0,0,0,0, 0,S3,0,0, 0,0,0,0, 0,0,0,0},
  /*k=6  z2  */ {0,0,0,0, 0,0,1.5f,0, 0,0,0,0, 0,0,0,0},
  /*k=7  xz  */ {0,0,0,0, 0,0,0,S3, 0,0,0,0, 0,0,0,0},
  /*k=8  x2  */ {0,0,0,0, 0,0,0,0, 0.8660254037844386f,0,0,0, 0,0,0,0},
  /*k=9  y2  */ {0,0,0,0, 0,0,0,0, -0.8660254037844386f,0,0,0, 0,0,0,0},
  /*k=10 x2y */ {0,0,0,0, 0,0,0,0, 0,3.f*S10_4,0,0, 0,0,0,0},
  /*k=11 y3  */ {0,0,0,0, 0,0,0,0, 0,-S10_4,0,0, 0,0,0,0},
  /*k=12 xyz */ {0,0,0,0, 0,0,0,0, 0,0,S15,0, 0,0,0,0},
  /*k=13 yz2 */ {0,0,0,0, 0,0,0,0, 0,0,0,5.f*S6_4, 0,0,0,0},
  /*k=14 z3  */ {0,0,0,0, 0,0,0,0, 0,0,0,0, 2.5f,0,0,0},
  /*k=15 xz2 */ {0,0,0,0, 0,0,0,0, 0,0,0,0, 0,5.f*S6_4,0,0},
  /*k=16 x2z */ {0,0,0,0, 0,0,0,0, 0,0,0,0, 0,0,S15_2,0},
  /*k=17 y2z */ {0,0,0,0, 0,0,0,0, 0,0,0,0, 0,0,-S15_2,0},
  /*k=18 x3  */ {0,0,0,0, 0,0,0,0, 0,0,0,0, 0,0,0,S10_4},
  /*k=19 xy2 */ {0,0,0,0, 0,0,0,0, 0,0,0,0, 0,0,0,-3.f*S10_4},
};

__device__ __forceinline__ float sph_mono(int k, float x, float y, float z) {
  switch (k) {
    case 0:  return 1.0f;
    case 1:  return y;
    case 2:  return z;
    case 3:  return x;
    case 4:  return x * y;
    case 5:  return y * z;
    case 6:  return z * z;
    case 7:  return x * z;
    case 8:  return x * x;
    case 9:  return y * y;
    case 10: return x * x * y;
    case 11: return y * y * y;
    case 12: return x * y * z;
    case 13: return y * z * z;
    case 14: return z * z * z;
    case 15: return x * z * z;
    case 16: return x * x * z;
    case 17: return y * y * z;
    case 18: return x * x * x;
    case 19: return x * y * y;
  }
  return 0.0f;
}

// ---------------------------------------------------------------------------
// Kernel A: init output (embedding gather + zero feature cols) and zero nbh ws
// ---------------------------------------------------------------------------
__global__ __launch_bounds__(256) void init_nodes_kernel(
    const int* __restrict__ Z, const float* __restrict__ emb,
    float* __restrict__ out, float* __restrict__ nbh, int N) {
  long long t = (long long)blockIdx.x * blockDim.x + threadIdx.x;
  if (t >= (long long)N * F_OUT) return;
  int n = (int)(t / F_OUT);
  int f = (int)(t % F_OUT);
  int zn = Z[n];
  float v = 0.0f;
  if (f < F_SCALAR && zn != 0) v = emb[zn * F_SCALAR + f];
  out[t] = v;
  if (f == F_OUT - 1) nbh[n] = 0.0f;   // fresh nbh accumulator each launch
}

// ---------------------------------------------------------------------------
// Kernel B: edge pass A — cosine cutoff -> atomic segment sum into nbh[i]
// ---------------------------------------------------------------------------
__global__ __launch_bounds__(256) void edge_nbh_kernel(
    const float* __restrict__ dr, const int* __restrict__ idx,
    float* __restrict__ nbh, int E) {
  long long e = (long long)blockIdx.x * blockDim.x + threadIdx.x;
  if (e >= E) return;
  int i = idx[e];
  int j = idx[(long long)E + e];
  if (i == j) return;
  float x = dr[3 * e + 0], y = dr[3 * e + 1], z = dr[3 * e + 2];
  float r = sqrtf(x * x + y * y + z * z);
  if (r >= R_MAX_F) return;
  float c = cosf((PI_F / R_MAX_F) * r);
  atomicAdd(nbh + i, 0.5f * (c + 1.0f));
}

// ---------------------------------------------------------------------------
// Kernel C: edge pass B — spherical harmonics via chained V_WMMA_F32_16X16X4_F32
// (16 edges per wave, tile mirrored across lane halves per ISA A-matrix layout),
// Bessel radial basis via sine recurrence; atomic scatter into out[:,128:160].
// ---------------------------------------------------------------------------
__global__ __launch_bounds__(256) void edge_features_kernel(
    const float* __restrict__ dr, const int* __restrict__ idx,
    const float* __restrict__ nbh, float* __restrict__ out, int E) {
  const int lane   = threadIdx.x & 31;
  const int wave   = threadIdx.x >> 5;
  const int sub    = lane & 15;          // edge-in-tile this lane mirrors
  const bool hiHalf = lane >= 16;
  const long long base = ((long long)blockIdx.x * 8 + wave) * 16;
  const long long e = base + sub;
  const bool valid = (e < E);

  int i_node = 0, j_node = 0;
  float x = 0.0f, y = 0.0f, z = 0.0f;
  if (valid) {
    i_node = idx[e];
    j_node = idx[(long long)E + e];
    x = dr[3 * e + 0];
    y = dr[3 * e + 1];
    z = dr[3 * e + 2];
  }
  float r      = sqrtf(x * x + y * y + z * z);
  float safe_r = fmaxf(r, 1e-12f);
  float inv_r  = 1.0f / safe_r;
  float ux = x * inv_r, uy = y * inv_r, uz = z * inv_r;

  float s1, c1;
  sincosf((PI_F / R_MAX_F) * r, &s1, &c1);

  bool  pm     = valid && (i_node != j_node);
  float cutoff = (pm && r < R_MAX_F) ? 0.5f * (c1 + 1.0f) : 0.0f;
  float nb     = valid ? nbh[i_node] : 0.0f;
  float pscale = pm ? (1.0f / (nb > 0.0f ? nb : 1.0f)) : 0.0f;

  // ---- sph = (pscale * Mono[16x20]) x Coef[20x16] as 5 chained 16x16x4 WMMAs
  v8f acc = {0.f, 0.f, 0.f, 0.f, 0.f, 0.f, 0.f, 0.f};
#pragma unroll
  for (int c = 0; c < 5; ++c) {
    float mlo0 = sph_mono(4 * c + 0, ux, uy, uz);
    float mlo1 = sph_mono(4 * c + 1, ux, uy, uz);
    float mhi0 = sph_mono(4 * c + 2, ux, uy, uz);
    float mhi1 = sph_mono(4 * c + 3, ux, uy, uz);
    v2f A, B;
    A.x = (hiHalf ? mhi0 : mlo0) * pscale;   // lanes 0-15: K=4c,4c+1 ; 16-31: K=4c+2,4c+3
    A.y = (hiHalf ? mhi1 : mlo1) * pscale;
    int kb = 4 * c + (hiHalf ? 2 : 0);
    B.x = SPH_COEF[kb][sub];                 // B rows mirror A half/K mapping
    B.y = SPH_COEF[kb + 1][sub];
    acc = __builtin_amdgcn_wmma_f32_16x16x4_f32(
        /*neg_a=*/false, A, /*neg_b=*/false, B,
        /*c_mod=*/(short)0, acc, /*reuse_a=*/false, /*reuse_b=*/false);
  }

  // ---- scatter spherical: D layout — vgpr v, lane L => row M = v + 8*(L>=16), col N = L&15
#pragma unroll
  for (int rrow = 0; rrow < 8; ++rrow) {
    int m = rrow + (hiHalf ? 8 : 0);
    int ti = __shfl(i_node, m, 32);          // segment id of that tile row
    long long em = base + m;
    float v = acc[rrow];
    if (em < E && v != 0.0f)
      atomicAdd(out + (long long)ti * F_OUT + F_SCALAR + sub, v);
  }

  // ---- radial basis (lanes 0-15 own the 16 tile edges uniquely)
  if (!hiHalf && cutoff > 0.0f) {
    float pref = SQRT_2_OVER_RMAX * inv_r * cutoff;
    float twoc = 2.0f * c1;
    float sm1 = 0.0f, s = s1;
    float* dst = out + (long long)i_node * F_OUT + F_SCALAR + 16;
#pragma unroll
    for (int n = 0; n < NUM_RADIAL; ++n) {
      float v = pref * s;
      if (v != 0.0f) atomicAdd(dst + n, v);
      float nx = twoc * s - sm1;             // sin((n+1)t)=2cos(t)sin(nt)-sin((n-1)t)
      sm1 = s; s = nx;
    }
  }
}

// ---------------------------------------------------------------------------
// Kernel D: apply node_mask to aggregated feature columns
// ---------------------------------------------------------------------------
__global__ __launch_bounds__(256) void mask_nodes_kernel(
    const int* __restrict__ Z, float* __restrict__ out, int N) {
  long long t = (long long)blockIdx.x * blockDim.x + threadIdx.x;
  if (t >= (long long)N * 32) return;
  int n = (int)(t >> 5);
  int k = (int)(t & 31);
  if (Z[n] == 0) out[(long long)n * F_OUT + F_SCALAR + k] = 0.0f;
}

extern "C" void kernel_launch(void* const* d_in, const int* in_sizes, int n_in,
                              void* d_out, int out_size, void* d_ws, size_t ws_size,
                              hipStream_t stream) {
  const float* dr  = (const float*)d_in[0];   // [E,3] f32
  const int*   Z   = (const int*)d_in[1];     // [N]   i32
  const int*   idx = (const int*)d_in[2];     // [2,E] i32
  const float* emb = (const float*)d_in[3];   // [119,128] f32
  float* out = (float*)d_out;                 // [N,160] f32
  float* nbh = (float*)d_ws;                  // N floats of scratch

  const int N = in_sizes[1];
  const int E = in_sizes[2] / 2;

  long long initT = (long long)N * F_OUT;
  init_nodes_kernel<<<(unsigned)((initT + 255) / 256), 256, 0, stream>>>(Z, emb, out, nbh, N);

  edge_nbh_kernel<<<(unsigned)((E + 255) / 256), 256, 0, stream>>>(dr, idx, nbh, E);

  long long tiles  = ((long long)E + 15) / 16;      // 16 edges per wave
  long long blocks = (tiles + 7) / 8;               // 8 waves per block
  edge_features_kernel<<<(unsigned)blocks, 256, 0, stream>>>(dr, idx, nbh, out, E);

  long long maskT = (long long)N * 32;
  mask_nodes_kernel<<<(unsigned)((maskT + 255) / 256), 256, 0, stream>>>(Z, out, N);
}